// train_postprocessor_48722109006113
// MI455X (gfx1250) — compile-verified
//
#include <hip/hip_runtime.h>
#include <math.h>

#define BATCH   256
#define NQ      10000
#define NCLS    10
#define NC      100000          // NQ * NCLS
#define K       15
#define BLK     256
#define NWAVE   (BLK / 32)
#define CHUNK   12500           // floats per TDM chunk (50 KB)
#define NCHUNK  8               // 8 * 12500 = 100000

typedef unsigned int u32;
typedef unsigned long long u64;
typedef u32 u32x4 __attribute__((ext_vector_type(4)));
typedef int i32x4 __attribute__((ext_vector_type(4)));
typedef int i32x8 __attribute__((ext_vector_type(8)));

#if defined(__HIP_DEVICE_COMPILE__) && __has_builtin(__builtin_amdgcn_tensor_load_to_lds) && __has_builtin(__builtin_amdgcn_s_wait_tensorcnt)
#define USE_TDM 1
#else
#define USE_TDM 0
#endif

#if USE_TDM
// Build a Tensor-DMA descriptor (D#) for a 1-row 2D tile of `nfloats` f32 and
// issue TENSOR_LOAD_TO_LDS. Bit packing per CDNA5 ISA ch.8.3/8.4.
__device__ __forceinline__ void tdm_issue(u32 lds_off, const float* gsrc, u32 nfloats) {
  u64 ga = (u64)(uintptr_t)gsrc;
  u32x4 g0;
  g0.x = 1u;                                            // count=1 (valid), user mode
  g0.y = lds_off;                                       // LDS byte address
  g0.z = (u32)(ga & 0xffffffffu);                       // global_addr[31:0]
  g0.w = (u32)((ga >> 32) & 0x01ffffffu) | (2u << 30);  // global_addr[56:32] | type=2
  i32x8 g1;
  g1[0] = (int)(2u << 16);                      // wg_mask=0, data_size=2 (4B), no flags
  g1[1] = (int)((nfloats & 0xffffu) << 16);     // abar_addr=0 | tensor_dim0[15:0]
  g1[2] = (int)((nfloats >> 16) | (1u << 16));  // tensor_dim0[31:16] | tensor_dim1[15:0]=1
  g1[3] = (int)((nfloats & 0xffffu) << 16);     // tensor_dim1[31:16]=0 | tile_dim0
  g1[4] = 1;                                    // tile_dim1=1 | tile_dim2=0
  g1[5] = (int)nfloats;                         // tensor_dim0_stride[31:0]
  g1[6] = 0;                                    // stride hi / tdim1_stride lo
  g1[7] = 0;
  i32x4 z4 = {0, 0, 0, 0};
#if __clang_major__ >= 23
  i32x8 z8 = {0, 0, 0, 0, 0, 0, 0, 0};
  __builtin_amdgcn_tensor_load_to_lds(g0, g1, z4, z4, z8, 0);
#else
  __builtin_amdgcn_tensor_load_to_lds(g0, g1, z4, z4, 0);
#endif
}
#endif

__device__ __forceinline__ float sigmoidf_dev(float x) {
  return 1.0f / (1.0f + expf(-x));
}

// (value desc, index asc) strict ordering — matches jax.lax.top_k tie rule.
__device__ __forceinline__ bool better(float v1, int i1, float v2, int i2) {
  return (v1 > v2) || ((v1 == v2) && (i1 < i2));
}

// Register-resident sorted top-K insert (strict >, so equal values keep the
// earlier = lower index, matching jax).
__device__ __forceinline__ void consider(float v, int idx, float (&bv)[K], int (&bi)[K]) {
  if (v > bv[K - 1]) {
    bv[K - 1] = v; bi[K - 1] = idx;
    #pragma unroll
    for (int s = K - 1; s >= 1; --s) {
      if (bv[s] > bv[s - 1]) {
        float tv = bv[s]; bv[s] = bv[s - 1]; bv[s - 1] = tv;
        int ti = bi[s]; bi[s] = bi[s - 1]; bi[s - 1] = ti;
      }
    }
  }
}

__global__ __launch_bounds__(BLK, 1)
void train_postprocessor_48722109006113_kernel(const float* __restrict__ logits,
                                               const float* __restrict__ boxes,
                                               float* __restrict__ out) {
  const int b = blockIdx.x;
  const int tid = threadIdx.x;
  const int lane = tid & 31;
  const int wid = tid >> 5;
  const float* Lg = logits + (size_t)b * NC;

  __shared__ alignas(16) float lbuf[2][CHUNK];   // 100 KB double buffer
  __shared__ float wvv[NWAVE];                   // per-wave merge partials
  __shared__ int   wvi[NWAVE];
  __shared__ float topv[K];
  __shared__ int   topi[K];
  __shared__ float rows[K][NCLS + 4];

  // Per-thread sorted top-K (registers).
  float bv[K]; int bi[K];
  #pragma unroll
  for (int s = 0; s < K; ++s) { bv[s] = -INFINITY; bi[s] = 0x7fffffff; }

#if USE_TDM
  // ---- Streamed scan: TDM double-buffered global->LDS, wave 0 drives DMA ----
  if (tid < 32) tdm_issue((u32)(uintptr_t)(void*)&lbuf[0][0], Lg, CHUNK);
  for (int c = 0; c < NCHUNK; ++c) {
    if (tid < 32) {
      if (c + 1 < NCHUNK) {
        tdm_issue((u32)(uintptr_t)(void*)&lbuf[(c + 1) & 1][0],
                  Lg + (size_t)(c + 1) * CHUNK, CHUNK);
        __builtin_amdgcn_s_wait_tensorcnt(1);   // chunk c complete (in-order)
      } else {
        __builtin_amdgcn_s_wait_tensorcnt(0);
      }
    }
    __syncthreads();                            // buffer c&1 ready for all waves
    const float4* p4 = reinterpret_cast<const float4*>(&lbuf[c & 1][0]);
    const int base = c * CHUNK;
    for (int j = tid * 4; j + 4 <= CHUNK; j += BLK * 4) {
      float4 q = p4[j >> 2];                    // ds_load_b128, conflict-free
      consider(q.x, base + j + 0, bv, bi);
      consider(q.y, base + j + 1, bv, bi);
      consider(q.z, base + j + 2, bv, bi);
      consider(q.w, base + j + 3, bv, bi);
    }
    __syncthreads();                            // done reading before buffer reuse
  }
#else
  // ---- Fallback: direct coalesced b128 global loads ----
  const float4* g4 = reinterpret_cast<const float4*>(Lg);
  for (int j = tid * 4; j + 4 <= NC; j += BLK * 4) {
    float4 q = g4[j >> 2];
    consider(q.x, j + 0, bv, bi);
    consider(q.y, j + 1, bv, bi);
    consider(q.z, j + 2, bv, bi);
    consider(q.w, j + 3, bv, bi);
  }
#endif

  // ---- K-way head merge: per-thread lists are sorted, so the global top-15
  //      falls out of 15 rounds of (wave shuffle argmax -> 8 partials -> scan).
  for (int r = 0; r < K; ++r) {
    // Candidate = current head of this thread's sorted list (pure registers).
    float v = bv[0]; int i = bi[0];
    // Wave32 argmax reduce, no barriers.
    #pragma unroll
    for (int off = 16; off > 0; off >>= 1) {
      float ov = __shfl_down(v, (unsigned)off, 32);
      int oi = __shfl_down(i, (unsigned)off, 32);
      if ((lane + off) < 32 && better(ov, oi, v, i)) { v = ov; i = oi; }
    }
    if (lane == 0) { wvv[wid] = v; wvi[wid] = i; }
    __syncthreads();
    // Every thread scans the 8 wave partials -> identical global winner.
    float wv = wvv[0]; int wi = wvi[0];
    #pragma unroll
    for (int t = 1; t < NWAVE; ++t) {
      if (better(wvv[t], wvi[t], wv, wi)) { wv = wvv[t]; wi = wvi[t]; }
    }
    if (tid == 0) { topv[r] = wv; topi[r] = wi; }
    // Owner pops its head (register shift, unrolled).
    if (bi[0] == wi) {
      #pragma unroll
      for (int s = 0; s < K - 1; ++s) { bv[s] = bv[s + 1]; bi[s] = bi[s + 1]; }
      bv[K - 1] = -INFINITY; bi[K - 1] = 0x7fffffff;
    }
    __syncthreads();                            // partials consumed before reuse
  }

  // ---- Gather rows, apply threshold gate, build [K,14] ----
  if (tid < K) {
    const float lv = topv[tid];
    const int fi = topi[tid];
    const float score = sigmoidf_dev(lv);
    const float use = (score >= 0.7f) ? 1.0f : 0.0f;
    const int q = fi / NCLS;
    const float* bx = boxes + ((size_t)b * NQ + q) * 4;
    const float* lg = logits + ((size_t)b * NQ + q) * NCLS;
    float r[NCLS + 4];
    #pragma unroll
    for (int j = 0; j < 4; ++j) r[j] = bx[j] * use;
    #pragma unroll
    for (int c = 0; c < NCLS; ++c) r[4 + c] = sigmoidf_dev(lg[c]) * use;
    #pragma unroll
    for (int m = 0; m < NCLS + 4; ++m) rows[tid][m] = r[m];
  }
  __syncthreads();

  // ---- Second top-k: rank the 15 rows by rows[:,0] desc (ties: earlier k first) ----
  if (tid < K) {
    const float key = rows[tid][0];
    int rank = 0;
    #pragma unroll
    for (int j = 0; j < K; ++j) {
      if (j == tid) continue;
      const float kj = rows[j][0];
      rank += (kj > key) || ((kj == key) && (j < tid));
    }
    float* o = out + ((size_t)b * K + rank) * (NCLS + 4);
    #pragma unroll
    for (int m = 0; m < NCLS + 4; ++m) o[m] = rows[tid][m];
  }
}

extern "C" void kernel_launch(void* const* d_in, const int* in_sizes, int n_in,
                              void* d_out, int out_size, void* d_ws, size_t ws_size,
                              hipStream_t stream) {
  (void)in_sizes; (void)n_in; (void)out_size; (void)d_ws; (void)ws_size;
  const float* logits = (const float*)d_in[0];  // [256,10000,10] f32
  const float* boxes  = (const float*)d_in[1];  // [256,10000,4]  f32
  float* out = (float*)d_out;                   // [256,15,14]    f32
  train_postprocessor_48722109006113_kernel<<<dim3(BATCH), dim3(BLK), 0, stream>>>(
      logits, boxes, out);
}